// MultiHeadSelfAttention_65661460021618
// MI455X (gfx1250) — compile-verified
//
#include <hip/hip_runtime.h>

typedef __attribute__((ext_vector_type(2))) float v2f;
typedef __attribute__((ext_vector_type(8))) float v8f;

#define NUM_HEADS 4
#define HEAD_DIM 32
#define D_MODEL 128
#define SEQ 8192
#define BATCH 8
#define QKV_N 384        // 3 * H * hd
#define POS_PER_WG 16
#define KSPLIT 8         // K-dimension split for the output GEMM
#define KCHUNK (SEQ / KSPLIT)   // 1024
#define NTILES (BATCH * 8 * 8)  // 512 16x16 output tiles

// ---------------------------------------------------------------------------
// Kernel 1: fused QKV projection (fp32 WMMA) + per-position 4x4 head-mixing
// attention. Writes att[b][s][h*32+hd] flat to workspace (32 MB), which is
// exactly the (B, D=128, S=8192) row-major view kernel 2 needs.
// ---------------------------------------------------------------------------
__global__ __launch_bounds__(128) void qkv_attn_kernel(
    const float* __restrict__ x, const float* __restrict__ Wqkv,
    const float* __restrict__ bqkv, float* __restrict__ att)
{
  __shared__ float xs[POS_PER_WG][D_MODEL];         // 8 KB
  __shared__ float qkvs[POS_PER_WG][QKV_N + 8];     // ~25 KB, padded

  const int tid  = threadIdx.x;
  const int wave = tid >> 5;
  const int lane = tid & 31;
  const int g0   = blockIdx.x * POS_PER_WG;         // flat position index b*S+s

  // ---- Phase A: stage x tile (16 x 128), coalesced ----
  for (int i = tid; i < POS_PER_WG * D_MODEL; i += 128) {
    int r = i >> 7, c = i & 127;
    xs[r][c] = x[(g0 + r) * D_MODEL + c];
  }
  __syncthreads();

  // ---- Phase B: QKV projection via V_WMMA_F32_16X16X4_F32 ----
  // Wave w owns columns [96w, 96w+96) = head w's {q|k|v}; 6 tiles of 16 cols.
  const int colBase = wave * 96;
  const int halfSel = lane >> 4;        // 0 = lanes 0-15, 1 = lanes 16-31
  const int lr      = lane & 15;

  v8f acc[6];
  #pragma unroll
  for (int t = 0; t < 6; ++t) {
    float bv = bqkv[colBase + t * 16 + lr];         // bias depends on col only
    acc[t] = { bv, bv, bv, bv, bv, bv, bv, bv };
  }

  for (int ks = 0; ks < D_MODEL; ks += 4) {
    const int ak = ks + (halfSel << 1);             // K=ak, ak+1 in this half
    v2f a; a.x = xs[lr][ak]; a.y = xs[lr][ak + 1];  // A: 16x4 tile of x
    #pragma unroll
    for (int t = 0; t < 6; ++t) {
      const int col = colBase + t * 16 + lr;
      v2f bf;                                       // B: 4x16 tile of W_qkv
      bf.x = Wqkv[ak * QKV_N + col];
      bf.y = Wqkv[(ak + 1) * QKV_N + col];
      acc[t] = __builtin_amdgcn_wmma_f32_16x16x4_f32(
          false, a, false, bf, (short)0, acc[t], false, false);
    }
  }

  // Spill qkv tile (16 x 384) to LDS in (pos, col) layout
  #pragma unroll
  for (int t = 0; t < 6; ++t) {
    const int col = colBase + t * 16 + lr;
    #pragma unroll
    for (int r = 0; r < 8; ++r)
      qkvs[r + halfSel * 8][col] = acc[t][r];
  }
  __syncthreads();

  // ---- Phase C: per-position HxH softmax attention ----
  // thread -> (pos 0..15, head 0..3, half-of-hd 0..1)
  {
    const int pos = tid >> 3;
    const int sub = tid & 7;
    const int h   = sub >> 1;
    const int d0  = (sub & 1) * 16;
    const float scale = 0.17677669529663687f;       // 32^-0.5

    float s[NUM_HEADS];
    #pragma unroll
    for (int j = 0; j < NUM_HEADS; ++j) {
      float acc_s = 0.f;
      for (int d = 0; d < HEAD_DIM; ++d)
        acc_s += qkvs[pos][h * 96 + d] * qkvs[pos][j * 96 + 32 + d];
      s[j] = acc_s * scale;
    }
    float m = s[0];
    #pragma unroll
    for (int j = 1; j < NUM_HEADS; ++j) m = fmaxf(m, s[j]);
    float denom = 0.f;
    #pragma unroll
    for (int j = 0; j < NUM_HEADS; ++j) { s[j] = __expf(s[j] - m); denom += s[j]; }
    const float inv = 1.f / denom;

    float* dst = att + (g0 + pos) * D_MODEL + h * HEAD_DIM + d0;
    for (int d = 0; d < 16; ++d) {
      float o = 0.f;
      #pragma unroll
      for (int j = 0; j < NUM_HEADS; ++j)
        o += s[j] * qkvs[pos][j * 96 + 64 + d0 + d];
      dst[d] = o * inv;
    }
  }
}

// ---------------------------------------------------------------------------
// Kernel 2a: partial GEMM. (B, 128, 8192) @ (8192, 128) with K split 8 ways:
// one wave per (tile, k-chunk) -> 4096 waves, 256 fp32 WMMA K-steps each.
// Partial 16x16 tiles written to workspace (deterministic two-pass reduce,
// no float atomics).
// ---------------------------------------------------------------------------
__global__ __launch_bounds__(32) void out_gemm_partial_kernel(
    const float* __restrict__ att, const float* __restrict__ Wout,
    float* __restrict__ partials)
{
  const int lane   = threadIdx.x & 31;
  const int id     = blockIdx.x;           // 0..4095
  const int tileId = id >> 3;              // 0..511
  const int chunk  = id & (KSPLIT - 1);    // 0..7
  const int b  = tileId >> 6;
  const int mt = (tileId >> 3) & 7;
  const int nt = tileId & 7;

  const int lr      = lane & 15;
  const int halfSel = lane >> 4;
  const int row = mt * 16 + lr;
  const int col = nt * 16 + lr;

  const float* A  = att + b * (SEQ * D_MODEL) + row * SEQ;  // reshape view
  const float* Bp = Wout + col;

  v8f c = { 0.f, 0.f, 0.f, 0.f, 0.f, 0.f, 0.f, 0.f };

  const int k0 = chunk * KCHUNK;
  #pragma unroll 4
  for (int k = k0; k < k0 + KCHUNK; k += 4) {
    const int kk = k + (halfSel << 1);
    v2f a = *(const v2f*)(A + kk);          // 8B-aligned (kk even)
    v2f bf;
    bf.x = Bp[kk * D_MODEL];
    bf.y = Bp[(kk + 1) * D_MODEL];
    c = __builtin_amdgcn_wmma_f32_16x16x4_f32(
        false, a, false, bf, (short)0, c, false, false);
  }

  float* dst = partials + (tileId * KSPLIT + chunk) * 256;
  #pragma unroll
  for (int r = 0; r < 8; ++r)
    dst[(r + halfSel * 8) * 16 + lr] = c[r];
}

// ---------------------------------------------------------------------------
// Kernel 2b: reduce the 8 K-chunk partials per tile (fixed order =
// deterministic), add bias, scatter to the (B, 128, 128) output.
// ---------------------------------------------------------------------------
__global__ __launch_bounds__(256) void out_reduce_kernel(
    const float* __restrict__ partials, const float* __restrict__ bout,
    float* __restrict__ out)
{
  const int tileId = blockIdx.x;           // 0..511
  const int tid    = threadIdx.x;          // 0..255 -> (rr, cc) in tile
  const int rr = tid >> 4;
  const int cc = tid & 15;
  const int b  = tileId >> 6;
  const int mt = (tileId >> 3) & 7;
  const int nt = tileId & 7;

  const float* src = partials + tileId * (KSPLIT * 256) + tid;
  float acc = bout[nt * 16 + cc];
  #pragma unroll
  for (int ch = 0; ch < KSPLIT; ++ch)
    acc += src[ch * 256];

  out[b * (D_MODEL * D_MODEL) + (mt * 16 + rr) * D_MODEL + nt * 16 + cc] = acc;
}

// ---------------------------------------------------------------------------
extern "C" void kernel_launch(void* const* d_in, const int* in_sizes, int n_in,
                              void* d_out, int out_size, void* d_ws, size_t ws_size,
                              hipStream_t stream) {
  const float* x    = (const float*)d_in[0];
  const float* Wqkv = (const float*)d_in[1];
  const float* bqkv = (const float*)d_in[2];
  const float* Wout = (const float*)d_in[3];
  const float* bout = (const float*)d_in[4];

  float* att      = (float*)d_ws;                          // 8*8192*128 = 32 MB
  float* partials = att + BATCH * SEQ * D_MODEL;           // 512*8*256 = 4 MB
  float* out      = (float*)d_out;                         // B*128*128

  qkv_attn_kernel<<<(BATCH * SEQ) / POS_PER_WG, 128, 0, stream>>>(x, Wqkv, bqkv, att);
  out_gemm_partial_kernel<<<NTILES * KSPLIT, 32, 0, stream>>>(att, Wout, partials);
  out_reduce_kernel<<<NTILES, 256, 0, stream>>>(partials, bout, out);
}